// Uncondition9RotR_24799141167112
// MI455X (gfx1250) — compile-verified
//
#include <hip/hip_runtime.h>
#include <stdint.h>

typedef float v2f  __attribute__((ext_vector_type(2)));
typedef float v8f  __attribute__((ext_vector_type(8)));
typedef unsigned int u32x4 __attribute__((ext_vector_type(4)));
typedef int   i32x4 __attribute__((ext_vector_type(4)));
typedef int   i32x8 __attribute__((ext_vector_type(8)));

#define MATS_PER_BLOCK 256
#define THREADS 256

// Build a 1-D (dword-granularity) TDM descriptor per CDNA5 ISA 8.3/8.4 and issue it.
// group0: [1:0]=count=1, [63:32]=lds_addr, [120:64]=global_addr, [127:126]=type=2
// group1: data_size=4B, tensor_dim0=tile_dim0=ndw, tensor_dim1=tile_dim1=1,
//         tensor_dim0_stride=ndw, everything else 0.
__device__ __forceinline__ void tdm_issue(bool store, uint32_t lds_addr,
                                          uint64_t gaddr, uint32_t ndw) {
  u32x4 g0;
  g0.x = 1u;                                   // count=1 (valid), user mode
  g0.y = lds_addr;                             // LDS byte address
  g0.z = (uint32_t)(gaddr & 0xFFFFFFFFu);      // global_addr[31:0]
  g0.w = ((uint32_t)(gaddr >> 32) & 0x1FFFFFFu) | (2u << 30); // ga[56:32] | type=2
  i32x8 g1;
  g1[0] = (int)(2u << 16);                     // workgroup_mask=0 | data_size=2 (4 B)
  g1[1] = (int)((ndw & 0xFFFFu) << 16);        // abar_addr=0 | tensor_dim0[15:0]
  g1[2] = (int)(((ndw >> 16) & 0xFFFFu) | (1u << 16)); // td0[31:16] | tensor_dim1[15:0]=1
  g1[3] = (int)((ndw & 0xFFFFu) << 16);        // tensor_dim1[31:16]=0 | tile_dim0=ndw
  g1[4] = (int)1;                              // tile_dim1=1 | tile_dim2=0
  g1[5] = (int)ndw;                            // tensor_dim0_stride[31:0]
  g1[6] = 0;                                   // stride0[47:32] | stride1[15:0]
  g1[7] = 0;
  i32x4 z4 = {0, 0, 0, 0};
  i32x8 z8 = {0, 0, 0, 0, 0, 0, 0, 0};
  if (store) __builtin_amdgcn_tensor_store_from_lds(g0, g1, z4, z4, z8, 0);
  else       __builtin_amdgcn_tensor_load_to_lds(g0, g1, z4, z4, z8, 0);
}

__global__ void __launch_bounds__(THREADS)
polar3x3_kernel(const float* __restrict__ rot, const float* __restrict__ mat,
                float* __restrict__ out, float* __restrict__ logdet, int N) {
  __shared__ float ldsIn [MATS_PER_BLOCK * 9];   // 9216 B: R tiles, row-major
  __shared__ float ldsOut[MATS_PER_BLOCK * 9];   // 9216 B: results

  const int tid    = threadIdx.x;
  const int block0 = blockIdx.x * MATS_PER_BLOCK;
  int mvalid = N - block0;
  if (mvalid > MATS_PER_BLOCK) mvalid = MATS_PER_BLOCK;
  if (mvalid <= 0) return;                        // uniform across block
  const uint32_t ndw = (uint32_t)mvalid * 9u;     // valid dwords in tile

  // ---- Phase 1: TDM bulk load global -> LDS (wave 0 issues; TENSORcnt wait) ----
  if (tid < 32) {
    tdm_issue(false, (uint32_t)(uintptr_t)&ldsIn[0],
              (uint64_t)(uintptr_t)(rot + (size_t)block0 * 9), ndw);
    __builtin_amdgcn_s_wait_tensorcnt(0);
  }
  __syncthreads();

  // ---- Phase 2: new_mat = R @ mat via V_WMMA_F32_16X16X4_F32 ----
  // Treat R as (3N x 3) row-major; each WMMA multiplies 16 independent rows by mat.
  const int lane = tid & 31;
  const int wave = tid >> 5;
  const int half = lane >> 4;       // 0: K=0/1 (A), rows 0-7 (D); 1: K=2/3, rows 8-15
  const int nn   = lane & 15;
  // B operand (4x16, K per VGPR mirrored like A): B[k][n] = mat[k*3+n] for k,n<3 else 0
  v2f Bv;
  Bv.x = (nn < 3) ? mat[(half ? 2 : 0) * 3 + nn] : 0.0f;   // K = 0 or 2
  Bv.y = (!half && nn < 3) ? mat[3 + nn] : 0.0f;           // K = 1; K=3 padded 0
  const int rowWave = wave * 96;    // each wave owns 32 matrices = 96 rows
#pragma unroll
  for (int w = 0; w < 6; ++w) {
    const int rowBase = rowWave + w * 16;
    const int rA = rowBase + nn;                  // this lane's source row
    v2f Av;
    Av.x = ldsIn[rA * 3 + (half ? 2 : 0)];        // K = 0 or 2
    Av.y = half ? 0.0f : ldsIn[rA * 3 + 1];       // K = 1; K=3 padded 0
    v8f C = {};
    v8f D = __builtin_amdgcn_wmma_f32_16x16x4_f32(
        false, Av, false, Bv, (short)0, C, false, false);
    // D: vgpr i -> (M = i + 8*half, N = nn). Only columns 0..2 are real.
    if (nn < 3) {
      const int rOut = rowBase + (half ? 8 : 0);
#pragma unroll
      for (int i = 0; i < 8; ++i)
        ldsOut[(rOut + i) * 3 + nn] = D[i];
    }
  }
  __syncthreads();

  // ---- Phase 3: polar factor via det-scaled Newton: X <- a*X + b*cof(X) ----
  if (tid < mvalid) {
    float x[9];
#pragma unroll
    for (int j = 0; j < 9; ++j) x[j] = ldsOut[tid * 9 + j];
#pragma unroll
    for (int it = 0; it < 6; ++it) {
      float c0 = x[4]*x[8] - x[5]*x[7];
      float c1 = x[5]*x[6] - x[3]*x[8];
      float c2 = x[3]*x[7] - x[4]*x[6];
      float c3 = x[2]*x[7] - x[1]*x[8];
      float c4 = x[0]*x[8] - x[2]*x[6];
      float c5 = x[1]*x[6] - x[0]*x[7];
      float c6 = x[1]*x[5] - x[2]*x[4];
      float c7 = x[2]*x[3] - x[0]*x[5];
      float c8 = x[0]*x[4] - x[1]*x[3];
      float det = x[0]*c0 + x[1]*c1 + x[2]*c2;   // X^-T = cof/det
      float fa  = fmaxf(fabsf(det), 1e-25f);
      float ds  = copysignf(fa, det);
      float ga, gb;
      if (it < 4) {                               // determinant scaling g=|det|^(-1/3)
        float g = __builtin_amdgcn_exp2f(-0.33333334f * __builtin_amdgcn_logf(fa));
        ga = 0.5f * g;
        gb = 0.5f * __builtin_amdgcn_rcpf(g * ds);
      } else {                                    // unscaled quadratic finish
        ga = 0.5f;
        gb = 0.5f * __builtin_amdgcn_rcpf(ds);
      }
      x[0] = ga*x[0] + gb*c0;  x[1] = ga*x[1] + gb*c1;  x[2] = ga*x[2] + gb*c2;
      x[3] = ga*x[3] + gb*c3;  x[4] = ga*x[4] + gb*c4;  x[5] = ga*x[5] + gb*c5;
      x[6] = ga*x[6] + gb*c6;  x[7] = ga*x[7] + gb*c7;  x[8] = ga*x[8] + gb*c8;
    }
#pragma unroll
    for (int j = 0; j < 9; ++j) ldsOut[tid * 9 + j] = x[j];
  }
  __syncthreads();

  // ---- Phase 4: TDM bulk store LDS -> global, plus logdet = 0 ----
  if (tid < 32) {
    tdm_issue(true, (uint32_t)(uintptr_t)&ldsOut[0],
              (uint64_t)(uintptr_t)(out + (size_t)block0 * 9), ndw);
    __builtin_amdgcn_s_wait_tensorcnt(0);
  }
  if (tid < mvalid) logdet[block0 + tid] = 0.0f;
}

extern "C" void kernel_launch(void* const* d_in, const int* in_sizes, int n_in,
                              void* d_out, int out_size, void* d_ws, size_t ws_size,
                              hipStream_t stream) {
  (void)n_in; (void)out_size; (void)d_ws; (void)ws_size;
  const float* rot = (const float*)d_in[0];
  const float* mat = (const float*)d_in[1];
  const int N = in_sizes[0] / 9;
  float* out    = (float*)d_out;
  float* logdet = out + (size_t)N * 9;
  const int blocks = (N + MATS_PER_BLOCK - 1) / MATS_PER_BLOCK;
  polar3x3_kernel<<<blocks, THREADS, 0, stream>>>(rot, mat, out, logdet, N);
}